// GCNModel_72619307041204
// MI455X (gfx1250) — compile-verified
//
#include <hip/hip_runtime.h>
#include <hip/hip_bf16.h>

// GCN 2-layer forward for MI455X (gfx1250, wave32).
// GEMMs use V_WMMA_F32_16X16X4_F32 (fp32-exact matrix path);
// edge aggregation uses native global_atomic_add_f32 streaming.

#define N_NODES 50000
#define N_EDGES 500000
#define DHID    128

typedef __attribute__((ext_vector_type(2))) float v2f;
typedef __attribute__((ext_vector_type(8))) float v8f;

// ---------------------------------------------------------------------------
// degree accumulation: deg[dst] += w[e]
__global__ __launch_bounds__(256) void deg_kernel(const long long* __restrict__ dsts,
                                                  const float* __restrict__ ew,
                                                  float* __restrict__ deg) {
    unsigned e = blockIdx.x * 256u + threadIdx.x;
    if (e >= N_EDGES) return;
    int d = (int)dsts[e];
    __hip_atomic_fetch_add(&deg[d], ew[e], __ATOMIC_RELAXED, __HIP_MEMORY_SCOPE_AGENT);
}

// dinv[i] = rsqrt(deg[i] + 1)   (deg+1 >= 1 always, matches reference where())
__global__ __launch_bounds__(256) void dinv_kernel(float* __restrict__ deg_to_dinv) {
    unsigned i = blockIdx.x * 256u + threadIdx.x;
    if (i >= N_NODES) return;
    deg_to_dinv[i] = __frsqrt_rn(deg_to_dinv[i] + 1.0f);
}

// norm[e] = dinv[src] * w[e] * dinv[dst]
__global__ __launch_bounds__(256) void norm_kernel(const long long* __restrict__ srcs,
                                                   const long long* __restrict__ dsts,
                                                   const float* __restrict__ ew,
                                                   const float* __restrict__ dinv,
                                                   float* __restrict__ norm) {
    unsigned e = blockIdx.x * 256u + threadIdx.x;
    if (e >= N_EDGES) return;
    int s = (int)srcs[e];
    int d = (int)dsts[e];
    norm[e] = dinv[s] * ew[e] * dinv[d];
}

// ---------------------------------------------------------------------------
// Y[M,128] = X[M,128] @ W[128,128] using fp32 WMMA 16x16x4.
// Block = 256 threads = 8 waves; block owns 16 rows, wave w owns cols w*16..w*16+15.
// A frag (16x4 f32, 2 VGPRs): lane L -> row M = L&15 ; K pair base = 2*(L>>4).
// B frag (4x16 f32, 2 VGPRs): lane L -> col N = L&15 ; K pair base = 2*(L>>4).
// C/D (16x16 f32, 8 VGPRs): vgpr r -> M = r + 8*(L>>4), N = L&15.
__global__ __launch_bounds__(256) void gemm_xw_wmma(const float* __restrict__ X,
                                                    const float* __restrict__ W,
                                                    float* __restrict__ Y) {
    const int wave    = threadIdx.x >> 5;       // 0..7 column tile
    const int lane    = threadIdx.x & 31;
    const int rowbase = blockIdx.x * 16;
    const int m       = lane & 15;
    const int khalf   = (lane >> 4) << 1;       // 0 or 2
    const int col     = (wave << 4) + m;

    const float* __restrict__ xrow = X + (size_t)(rowbase + m) * DHID;
    v8f acc = {};

#pragma unroll 4
    for (int k0 = 0; k0 < DHID; k0 += 4) {
        const int k = k0 + khalf;
        v2f a, b;
        a.x = xrow[k];
        a.y = xrow[k + 1];
        b.x = W[(size_t)k * DHID + col];
        b.y = W[(size_t)(k + 1) * DHID + col];
        acc = __builtin_amdgcn_wmma_f32_16x16x4_f32(false, a, false, b,
                                                    (short)0, acc, false, false);
    }

    const int mofs = (lane >> 4) * 8;
#pragma unroll
    for (int r = 0; r < 8; ++r) {
        Y[(size_t)(rowbase + mofs + r) * DHID + col] = acc[r];
    }
}

// ---------------------------------------------------------------------------
// Edge message scatter: AGG[dst] += XW[src] * norm[e].
// One wave per edge; each lane handles 4 consecutive features (float4 gather).
__global__ __launch_bounds__(256) void aggregate_kernel(const float* __restrict__ XW,
                                                        const long long* __restrict__ srcs,
                                                        const long long* __restrict__ dsts,
                                                        const float* __restrict__ norm,
                                                        float* __restrict__ AGG) {
    unsigned gid = blockIdx.x * 256u + threadIdx.x;
    unsigned e   = gid >> 5;
    if (e >= N_EDGES) return;
    int f = (int)(gid & 31u) << 2;

    int   s  = (int)srcs[e];
    int   d  = (int)dsts[e];
    float nw = norm[e];

    float4 v = *(const float4*)(XW + (size_t)s * DHID + f);
    float* o = AGG + (size_t)d * DHID + f;
    __hip_atomic_fetch_add(o + 0, v.x * nw, __ATOMIC_RELAXED, __HIP_MEMORY_SCOPE_AGENT);
    __hip_atomic_fetch_add(o + 1, v.y * nw, __ATOMIC_RELAXED, __HIP_MEMORY_SCOPE_AGENT);
    __hip_atomic_fetch_add(o + 2, v.z * nw, __ATOMIC_RELAXED, __HIP_MEMORY_SCOPE_AGENT);
    __hip_atomic_fetch_add(o + 3, v.w * nw, __ATOMIC_RELAXED, __HIP_MEMORY_SCOPE_AGENT);
}

// Self-loop contribution + bias: AGG[n,f] += XW[n,f]*dinv[n]^2 + b[f]
__global__ __launch_bounds__(256) void self_bias_kernel(float* __restrict__ AGG,
                                                        const float* __restrict__ XW,
                                                        const float* __restrict__ dinv,
                                                        const float* __restrict__ bias) {
    unsigned gid = blockIdx.x * 256u + threadIdx.x;   // N_NODES*128 threads
    unsigned n = gid >> 7;
    unsigned f = gid & 127u;
    if (n >= N_NODES) return;
    float di = dinv[n];
    AGG[gid] += XW[gid] * (di * di) + bias[f];
}

// out[n] = sigmoid(dot(H[n,:], Wf) + bf) * 10 ; one wave per node
__global__ __launch_bounds__(256) void final_kernel(const float* __restrict__ H,
                                                    const float* __restrict__ Wf,
                                                    const float* __restrict__ bf,
                                                    float* __restrict__ out) {
    int node = blockIdx.x * 8 + (threadIdx.x >> 5);
    int lane = threadIdx.x & 31;
    if (node >= N_NODES) return;

    float4 h = *(const float4*)(H + (size_t)node * DHID + lane * 4);
    float4 w = *(const float4*)(Wf + lane * 4);
    float s = h.x * w.x + h.y * w.y + h.z * w.z + h.w * w.w;
#pragma unroll
    for (int off = 16; off > 0; off >>= 1)
        s += __shfl_xor(s, off, 32);

    if (lane == 0) {
        float z = s + bf[0];
        out[node] = 10.0f / (1.0f + __expf(-z));
    }
}

// ---------------------------------------------------------------------------
extern "C" void kernel_launch(void* const* d_in, const int* in_sizes, int n_in,
                              void* d_out, int out_size, void* d_ws, size_t ws_size,
                              hipStream_t stream) {
    const float*     x    = (const float*)d_in[0];
    const long long* eidx = (const long long*)d_in[1];   // jnp.int64 [2, E]
    const float*     ew   = (const float*)d_in[2];
    const float*     W1   = (const float*)d_in[3];
    const float*     b1   = (const float*)d_in[4];
    const float*     W2   = (const float*)d_in[5];
    const float*     b2   = (const float*)d_in[6];
    const float*     Wf   = (const float*)d_in[7];
    const float*     bf   = (const float*)d_in[8];
    float*           out  = (float*)d_out;

    const long long* srcs = eidx;
    const long long* dsts = eidx + N_EDGES;

    // workspace layout (256B aligned chunks)
    char* ws = (char*)d_ws;
    size_t off = 0;
    auto carve = [&](size_t bytes) {
        char* p = ws + off;
        off += (bytes + 255) & ~size_t(255);
        return p;
    };
    float* dinv = (float*)carve(sizeof(float) * N_NODES);
    float* norm = (float*)carve(sizeof(float) * N_EDGES);
    float* bufA = (float*)carve(sizeof(float) * (size_t)N_NODES * DHID);
    float* bufB = (float*)carve(sizeof(float) * (size_t)N_NODES * DHID);
    (void)ws_size; (void)in_sizes; (void)n_in; (void)out_size;

    const int eBlocks    = (N_EDGES + 255) / 256;
    const int nBlocks    = (N_NODES + 255) / 256;
    const int gemmBlocks = N_NODES / 16;                      // 3125
    const int aggBlocks  = (N_EDGES * 32) / 256;              // 62500
    const int sbBlocks   = (N_NODES * DHID) / 256;            // 25000
    const int finBlocks  = (N_NODES + 7) / 8;                 // 6250

    // --- graph normalization (once) ---
    hipMemsetAsync(dinv, 0, sizeof(float) * N_NODES, stream);
    deg_kernel<<<eBlocks, 256, 0, stream>>>(dsts, ew, dinv);
    dinv_kernel<<<nBlocks, 256, 0, stream>>>(dinv);
    norm_kernel<<<eBlocks, 256, 0, stream>>>(srcs, dsts, ew, dinv, norm);

    // --- layer 1: h1 = Agg(x @ W1) + self + b1 ---
    gemm_xw_wmma<<<gemmBlocks, 256, 0, stream>>>(x, W1, bufA);
    hipMemsetAsync(bufB, 0, sizeof(float) * (size_t)N_NODES * DHID, stream);
    aggregate_kernel<<<aggBlocks, 256, 0, stream>>>(bufA, srcs, dsts, norm, bufB);
    self_bias_kernel<<<sbBlocks, 256, 0, stream>>>(bufB, bufA, dinv, b1);

    // --- layer 2: h2 = Agg(h1 @ W2) + self + b2 ---
    gemm_xw_wmma<<<gemmBlocks, 256, 0, stream>>>(bufB, W2, bufA);
    hipMemsetAsync(bufB, 0, sizeof(float) * (size_t)N_NODES * DHID, stream);
    aggregate_kernel<<<aggBlocks, 256, 0, stream>>>(bufA, srcs, dsts, norm, bufB);
    self_bias_kernel<<<sbBlocks, 256, 0, stream>>>(bufB, bufA, dinv, b2);

    // --- readout ---
    final_kernel<<<finBlocks, 256, 0, stream>>>(bufB, Wf, bf, out);
}